// CustomGraphConvLayer_81252191306416
// MI455X (gfx1250) — compile-verified
//
#include <hip/hip_runtime.h>
#include <hip/hip_bf16.h>

typedef __attribute__((ext_vector_type(2))) float v2f;
typedef __attribute__((ext_vector_type(8))) float v8f;

#define D 128
#define LDS_PITCH 130   // 130 % 64 == 2 -> 16 lanes with row-stride hit distinct banks; keeps 8B alignment

// ---------------------------------------------------------------------------
// Kernel 0: zero the workspace (seg_sum [N*D] + counts [N])
// ---------------------------------------------------------------------------
__global__ void zero_ws(float* __restrict__ p, int n) {
    int stride = gridDim.x * blockDim.x;
    for (int i = blockIdx.x * blockDim.x + threadIdx.x; i < n; i += stride)
        p[i] = 0.0f;
}

// ---------------------------------------------------------------------------
// Kernel 1: edge gather + atomic scatter-add (segment sum + degree counts).
// One 32-lane wave slot per edge; each lane moves 4 consecutive floats.
// x[dst[e]] row load is fully coalesced (512B contiguous); the adds hit an
// L2-resident 25.6MB accumulator -> native global_atomic_add_f32.
// ---------------------------------------------------------------------------
__global__ __launch_bounds__(256)
void scatter_accum(const float* __restrict__ x,
                   const long long* __restrict__ src,
                   const long long* __restrict__ dst,
                   float* __restrict__ seg,
                   float* __restrict__ cnt,
                   int E) {
    int gid  = blockIdx.x * blockDim.x + threadIdx.x;
    int e    = gid >> 5;
    int lane = gid & 31;
    if (e >= E) return;
    int s = (int)src[e];
    int d = (int)dst[e];
    float4 v = ((const float4*)(x + (size_t)d * D))[lane];
    float* sp = seg + (size_t)s * D + lane * 4;
    unsafeAtomicAdd(sp + 0, v.x);
    unsafeAtomicAdd(sp + 1, v.y);
    unsafeAtomicAdd(sp + 2, v.z);
    unsafeAtomicAdd(sp + 3, v.w);
    if (lane == 0) unsafeAtomicAdd(cnt + s, 1.0f);
}

// ---------------------------------------------------------------------------
// Kernel 2: fused  relu(x@Ws^T + b_s + mask*(mean@Wn^T + b_n))
// One wave per 16-row tile; 8 waves/block. Both GEMMs accumulate into the
// SAME 8 v8f accumulators (v_wmma_f32_16x16x4_f32, K stepped 4 at a time).
// W_self / W_neighbor staged in LDS with pitch 130 (conflict-free B reads).
// ---------------------------------------------------------------------------
__global__ __launch_bounds__(256)
void fused_sage(const float* __restrict__ x,
                const float* __restrict__ seg,
                const float* __restrict__ cnt,
                const float* __restrict__ Ws,
                const float* __restrict__ bs,
                const float* __restrict__ Wn,
                const float* __restrict__ bn,
                float* __restrict__ out,
                int nrowtiles) {
    __shared__ float wlds[2 * D * LDS_PITCH];   // 133,120 bytes (<320KB/WGP)

    // Cooperative load of both 128x128 weight matrices into LDS.
    for (int m = 0; m < 2; ++m) {
        const float* W = m ? Wn : Ws;
        float* l = wlds + m * D * LDS_PITCH;
        for (int i = threadIdx.x; i < D * (D / 4); i += 256) {
            int r  = i >> 5;   // row (output feature)
            int c4 = i & 31;   // float4 within row (input feature / 4)
            float4 v = ((const float4*)(W + (size_t)r * D))[c4];
            float* dl = l + r * LDS_PITCH + c4 * 4;
            dl[0] = v.x; dl[1] = v.y; dl[2] = v.z; dl[3] = v.w;
        }
    }
    __syncthreads();

    int wave = threadIdx.x >> 5;
    int lane = threadIdx.x & 31;
    int tile = blockIdx.x * 8 + wave;
    if (tile >= nrowtiles) return;          // uniform per wave -> EXEC all-1s
    int rb   = tile * 16;
    int half = lane >> 4;                   // 0: K={k0,k0+1}; 1: K={k0+2,k0+3}
    int l16  = lane & 15;
    int arow = rb + l16;

    float c   = cnt[arow];
    float inv = 1.0f / fmaxf(c, 1.0f);      // mean = seg * inv (seg==0 if c==0)

    v8f acc[8];
    v8f zero = {};
#pragma unroll
    for (int i = 0; i < 8; ++i) acc[i] = zero;

    const float* xrow = x   + (size_t)arow * D + half * 2;
    const float* srow = seg + (size_t)arow * D + half * 2;

#pragma unroll 4
    for (int kk = 0; kk < D / 4; ++kk) {
        int k0 = kk * 4;
        // A fragments (ISA 16x4 f32 layout): one b64 load each
        v2f ax = *(const v2f*)(xrow + k0);
        v2f am = (*(const v2f*)(srow + k0)) * inv;
#pragma unroll
        for (int ct = 0; ct < 8; ++ct) {
            int brow = ct * 16 + l16;       // output feature (row of W)
            const float* bp = wlds + brow * LDS_PITCH + k0 + half * 2;
            v2f bS = *(const v2f*)bp;                         // W_self^T frag
            v2f bN = *(const v2f*)(bp + D * LDS_PITCH);       // W_neigh^T frag
            acc[ct] = __builtin_amdgcn_wmma_f32_16x16x4_f32(
                false, ax, false, bS, (short)0, acc[ct], false, false);
            acc[ct] = __builtin_amdgcn_wmma_f32_16x16x4_f32(
                false, am, false, bN, (short)0, acc[ct], false, false);
        }
    }

    // Epilogue: bias + neighbor-bias mask (count>0) + ReLU, per C/D layout:
    // VGPR r holds row M=r (lanes 0-15) / M=r+8 (lanes 16-31), col N=lane%16.
    float nmask[8];
#pragma unroll
    for (int r = 0; r < 8; ++r) {
        int row = rb + r + half * 8;
        nmask[r] = (cnt[row] > 0.0f) ? 1.0f : 0.0f;
    }
#pragma unroll
    for (int ct = 0; ct < 8; ++ct) {
        int col = ct * 16 + l16;
        float bsv = bs[col];
        float bnv = bn[col];
#pragma unroll
        for (int r = 0; r < 8; ++r) {
            int row = rb + r + half * 8;
            float v = acc[ct][r] + bsv + nmask[r] * bnv;
            out[(size_t)row * D + col] = fmaxf(v, 0.0f);
        }
    }
}

// ---------------------------------------------------------------------------
extern "C" void kernel_launch(void* const* d_in, const int* in_sizes, int n_in,
                              void* d_out, int out_size, void* d_ws, size_t ws_size,
                              hipStream_t stream) {
    const float*     x    = (const float*)d_in[0];
    const long long* ei   = (const long long*)d_in[1];
    const float*     Ws   = (const float*)d_in[2];
    const float*     bs   = (const float*)d_in[3];
    const float*     Wn   = (const float*)d_in[4];
    const float*     bn   = (const float*)d_in[5];
    float*           out  = (float*)d_out;

    const int N = in_sizes[0] / D;          // 50000
    const int E = in_sizes[1] / 2;          // 800000

    float* seg = (float*)d_ws;              // [N*D]
    float* cnt = seg + (size_t)N * D;       // [N]
    const long long* src = ei;              // edge_index[0]
    const long long* dst = ei + E;          // edge_index[1]

    // 1) zero workspace
    int nz = N * D + N;
    zero_ws<<<2048, 256, 0, stream>>>(seg, nz);

    // 2) gather + atomic segment-sum
    long long tot = (long long)E * 32;
    int blocks1 = (int)((tot + 255) / 256);
    scatter_accum<<<blocks1, 256, 0, stream>>>(x, src, dst, seg, cnt, E);

    // 3) fused dual-GEMM + bias + mask + relu via f32 WMMA
    int nrowtiles = N / 16;                 // 3125 (N divisible by 16)
    int blocks2 = (nrowtiles + 7) / 8;
    fused_sage<<<blocks2, 256, 0, stream>>>(x, seg, cnt, Ws, bs, Wn, bn, out, nrowtiles);
}